// StokenAttention_72035191488647
// MI455X (gfx1250) — compile-verified
//
#include <hip/hip_runtime.h>

// StokenAttention for MI455X (gfx1250, wave32).
// Memory-bound problem (~9 FLOP/byte): FP32 WMMA 16x16x4 keeps full precision,
// all data movement staged through LDS. Pixel staging uses the CDNA5 async
// global->LDS path (GLOBAL_LOAD_ASYNC_TO_LDS_B128 / ASYNCcnt), avoiding the
// VGPR round-trip + ds_store of a synchronous copy.

typedef float v2f __attribute__((ext_vector_type(2)));
typedef float v8f __attribute__((ext_vector_type(8)));
// exact pointee type of the async-load builtin (per hipcc diagnostic):
typedef int async_b128_t __attribute__((vector_size(4 * sizeof(int))));

#define NB     4
#define NC     384
#define NH     224
#define NW     224
#define HH     28
#define WW     28
#define NM     (HH * WW)     // 784 patches
#define NL     64            // pixels per patch
#define HP     8
#define WP     8
#define SCALEF 0.051031036307982884f   // 384^-0.5
#define FEPS   1e-12f
#define CHUNK  96            // channels staged in LDS at a time (4 chunks)
#define NCHUNK 4
#define PSTR   68            // pixT row stride (16B-aligned float4 stores, low bank conflict)
#define SSTR   17            // stu row stride  (conflict-free A-fragment reads)
#define ASTR   68            // afft row stride

#if defined(__has_builtin)
#if __has_builtin(__builtin_amdgcn_global_load_async_to_lds_b128)
#define USE_ASYNC_LDS 1
#endif
#endif
#ifndef USE_ASYNC_LDS
#define USE_ASYNC_LDS 0
#endif

#define AS_GLOBAL __attribute__((address_space(1)))
#define AS_LOCAL  __attribute__((address_space(3)))

static __device__ __forceinline__ void wait_async0() {
#if USE_ASYNC_LDS
#if __has_builtin(__builtin_amdgcn_s_wait_asynccnt)
  __builtin_amdgcn_s_wait_asynccnt(0);
#else
  asm volatile("s_wait_asynccnt 0x0" ::: "memory");
#endif
#endif
}

// D = A(16x4) * B(4x16) + C, fp32. Lane mapping per CDNA5 ISA 7.12.2.
static __device__ __forceinline__ v8f wmma4(v2f a, v2f b, v8f c) {
  return __builtin_amdgcn_wmma_f32_16x16x4_f32(false, a, false, b, (short)0, c,
                                               false, false);
}

// Stage one 96-channel slab of this patch's 64 pixels into LDS as pixT[c][l].
// 16B-aligned 128-bit transfers (patch cols are multiples of 4 within W=224).
// Async path: direct global->LDS DMA, waited with ASYNCcnt before the barrier.
static __device__ __forceinline__ void stage_pix_chunk(
    const float* __restrict__ x, float (*pixT)[PSTR],
    int b, int p, int q, int ch, int tid) {
  for (int j = tid; j < CHUNK * 16; j += 128) {
    int cl = j >> 4;            // chunk-local channel
    int l4 = j & 15;            // which float4 of the 8x8 patch
    int c  = ch * CHUNK + cl;
    int r  = l4 >> 1;
    int c4 = (l4 & 1) * 4;
    size_t gi = (((size_t)(b * NC + c) * NH + p * HP + r) * NW + q * WP + c4);
    float* dst = &pixT[cl][r * 8 + c4];
#if USE_ASYNC_LDS
    __builtin_amdgcn_global_load_async_to_lds_b128(
        (AS_GLOBAL async_b128_t*)(x + gi),
        (AS_LOCAL async_b128_t*)dst, 0, 0);
#else
    *reinterpret_cast<float4*>(dst) = *reinterpret_cast<const float4*>(x + gi);
#endif
  }
  wait_async0();   // this wave's async LDS writes landed; barrier covers others
}

// ---------------------------------------------------------------------------
// K1: initial super-tokens = 8x8 average pool, stored channel-last (B,28,28,C)
// so later 9-neighbor gathers are contiguous.
// ---------------------------------------------------------------------------
__global__ __launch_bounds__(128) void k_avgpool(const float* __restrict__ x,
                                                 float* __restrict__ st0) {
  __shared__ float pixT[CHUNK][PSTR];
  int tid = threadIdx.x;
  int bm  = blockIdx.x;
  int b = bm / NM, m = bm % NM;
  int p = m / WW, q = m % WW;
  for (int ch = 0; ch < NCHUNK; ++ch) {
    __syncthreads();
    stage_pix_chunk(x, pixT, b, p, q, ch, tid);
    __syncthreads();
    if (tid < CHUNK) {
      float s = 0.f;
#pragma unroll
      for (int l = 0; l < NL; ++l) s += pixT[tid][l];
      st0[(size_t)bm * NC + ch * CHUNK + tid] = s * (1.f / 64.f);
    }
  }
}

// ---------------------------------------------------------------------------
// K2a: iteration 0. Per patch:
//   logits(64x16) = pixels(64x384) @ st_unf(384x16)   [WMMA, K chunked]
//   affinity = softmax(logits * scale) over n<9
//   st(384x16) = pixels^T @ affinity                  [WMMA]
// st stored as (B,M,9,C) for coalesced stores + coalesced fold reads.
// ---------------------------------------------------------------------------
__global__ __launch_bounds__(128) void k_iter0(
    const float* __restrict__ x, const float* __restrict__ st0,
    float* __restrict__ st_ws, float* __restrict__ affsum_ws) {
  __shared__ float pixT[CHUNK][PSTR];   // pixels^T slab: [channel][pixel]
  __shared__ float stu[NC][SSTR];       // st_unf: [channel][neighbor<16]
  __shared__ float aff[NL][16];         // logits -> affinity: [pixel][neighbor]

  int tid = threadIdx.x;
  int bm  = blockIdx.x;
  int b = bm / NM, m = bm % NM;
  int p = m / WW, q = m % WW;
  int lane = tid & 31, wave = tid >> 5;
  int l15 = lane & 15, half = lane >> 4;

  // Gather 9 neighboring super-tokens (zero-padded at borders). Contiguous in c.
  for (int j = tid; j < 9 * (NC / 4); j += 128) {
    int n  = j / (NC / 4);
    int c4 = (j % (NC / 4)) * 4;
    int pp = p + n / 3 - 1, qq = q + n % 3 - 1;
    float4 v = make_float4(0.f, 0.f, 0.f, 0.f);
    if (pp >= 0 && pp < HH && qq >= 0 && qq < WW)
      v = *reinterpret_cast<const float4*>(
          st0 + ((size_t)(b * NM + pp * WW + qq) * NC + c4));
    stu[c4 + 0][n] = v.x; stu[c4 + 1][n] = v.y;
    stu[c4 + 2][n] = v.z; stu[c4 + 3][n] = v.w;
  }
  // (stu cols 9..15 left untouched: they only feed D columns we never read)

  // --- Phase 1: logits. Each wave owns one 16-row pixel tile (4 waves x 16 = 64).
  v8f acc = {0.f, 0.f, 0.f, 0.f, 0.f, 0.f, 0.f, 0.f};
  int m0 = wave * 16;
  for (int ch = 0; ch < NCHUNK; ++ch) {
    __syncthreads();
    stage_pix_chunk(x, pixT, b, p, q, ch, tid);
    __syncthreads();
    for (int k0 = 0; k0 < CHUNK; k0 += 4) {
      int kk = k0 + 2 * half;
      v2f a;  a.x = pixT[kk][m0 + l15];      a.y = pixT[kk + 1][m0 + l15];
      int kg = ch * CHUNK + kk;
      v2f bb; bb.x = stu[kg][l15];           bb.y = stu[kg + 1][l15];
      acc = wmma4(a, bb, acc);
    }
  }
#pragma unroll
  for (int r = 0; r < 8; ++r) aff[m0 + 8 * half + r][l15] = acc[r];
  __syncthreads();

  // --- Phase 2: row softmax over n<9 (thread per pixel row).
  if (tid < NL) {
    float v[9]; float mx = -1e30f;
#pragma unroll
    for (int n = 0; n < 9; ++n) { v[n] = aff[tid][n]; mx = fmaxf(mx, v[n]); }
    float s = 0.f;
#pragma unroll
    for (int n = 0; n < 9; ++n) { v[n] = __expf((v[n] - mx) * SCALEF); s += v[n]; }
    float inv = 1.f / s;
#pragma unroll
    for (int n = 0; n < 9; ++n) aff[tid][n] = v[n] * inv;
#pragma unroll
    for (int n = 9; n < 16; ++n) aff[tid][n] = 0.f;
  }
  __syncthreads();
  if (tid < 9) {                         // affinity column sums for fold norm
    float s = 0.f;
    for (int l = 0; l < NL; ++l) s += aff[l][tid];
    affsum_ws[(size_t)bm * 9 + tid] = s;
  }

  // --- Phase 3: st = pixels^T @ affinity. 6 channel tiles per chunk over 4 waves.
  for (int ch = 0; ch < NCHUNK; ++ch) {
    __syncthreads();
    stage_pix_chunk(x, pixT, b, p, q, ch, tid);
    __syncthreads();
    for (int t = wave; t < CHUNK / 16; t += 4) {
      int c0 = t * 16;
      v8f a2 = {0.f, 0.f, 0.f, 0.f, 0.f, 0.f, 0.f, 0.f};
#pragma unroll
      for (int k0 = 0; k0 < NL; k0 += 4) {
        int kk = k0 + 2 * half;
        v2f a;  a.x = pixT[c0 + l15][kk];  a.y = pixT[c0 + l15][kk + 1];
        v2f bb; bb.x = aff[kk][l15];       bb.y = aff[kk + 1][l15];
        a2 = wmma4(a, bb, a2);
      }
      if (l15 < 9) {                      // D rows = channels: 8 contiguous floats
        size_t base = ((size_t)bm * 9 + l15) * NC + ch * CHUNK + c0 + 8 * half;
        *reinterpret_cast<float4*>(st_ws + base) =
            make_float4(a2[0], a2[1], a2[2], a2[3]);
        *reinterpret_cast<float4*>(st_ws + base + 4) =
            make_float4(a2[4], a2[5], a2[6], a2[7]);
      }
    }
  }
}

// ---------------------------------------------------------------------------
// K2b: fold norm = eps + scatter-add of affinity sums (gather form, deterministic)
// ---------------------------------------------------------------------------
__global__ void k_norm(const float* __restrict__ affsum_ws,
                       float* __restrict__ norm_ws) {
  int idx = blockIdx.x * blockDim.x + threadIdx.x;
  if (idx >= NB * NM) return;
  int b = idx / NM, m = idx % NM;
  int p = m / WW, q = m % WW;
  float s = FEPS;
#pragma unroll
  for (int n = 0; n < 9; ++n) {
    int sp = p + 1 - n / 3, sq = q + 1 - n % 3;
    if (sp >= 0 && sp < HH && sq >= 0 && sq < WW)
      s += affsum_ws[(size_t)(b * NM + sp * WW + sq) * 9 + n];
  }
  norm_ws[idx] = s;
}

// ---------------------------------------------------------------------------
// K2c: refined super-tokens: gather-fold st contributions, divide by norm.
// Consecutive threads = consecutive channels -> fully coalesced.
// ---------------------------------------------------------------------------
__global__ void k_fold(const float* __restrict__ st_ws,
                       const float* __restrict__ norm_ws,
                       float* __restrict__ st1) {
  int idx = blockIdx.x * blockDim.x + threadIdx.x;     // exactly B*M*C threads
  int c = idx % NC;
  int t = idx / NC;
  int q = t % WW; t /= WW;
  int p = t % HH; int b = t / HH;
  float s = 0.f;
#pragma unroll
  for (int n = 0; n < 9; ++n) {
    int sp = p + 1 - n / 3, sq = q + 1 - n % 3;
    if (sp >= 0 && sp < HH && sq >= 0 && sq < WW)
      s += st_ws[((size_t)(b * NM + sp * WW + sq) * 9 + n) * NC + c];
  }
  st1[idx] = s / norm_ws[b * NM + p * WW + q];
}

// ---------------------------------------------------------------------------
// K3: final iteration: affinity from refined tokens, then
//   out(c,l) = st_unf(384x16) @ affinity^T(16x64)   [WMMA, K=16 zero-padded]
// ---------------------------------------------------------------------------
__global__ __launch_bounds__(128) void k_final(
    const float* __restrict__ x, const float* __restrict__ st1,
    float* __restrict__ out) {
  __shared__ float pixT[CHUNK][PSTR];
  __shared__ float stu[NC][SSTR];
  __shared__ float afft[16][ASTR];      // affinity^T: [neighbor][pixel]

  int tid = threadIdx.x;
  int bm  = blockIdx.x;
  int b = bm / NM, m = bm % NM;
  int p = m / WW, q = m % WW;
  int lane = tid & 31, wave = tid >> 5;
  int l15 = lane & 15, half = lane >> 4;

  // st_unf gather; cols 9..15 zeroed (they are the K-padding of the final GEMM).
  for (int j = tid; j < 9 * (NC / 4); j += 128) {
    int n  = j / (NC / 4);
    int c4 = (j % (NC / 4)) * 4;
    int pp = p + n / 3 - 1, qq = q + n % 3 - 1;
    float4 v = make_float4(0.f, 0.f, 0.f, 0.f);
    if (pp >= 0 && pp < HH && qq >= 0 && qq < WW)
      v = *reinterpret_cast<const float4*>(
          st1 + ((size_t)(b * NM + pp * WW + qq) * NC + c4));
    stu[c4 + 0][n] = v.x; stu[c4 + 1][n] = v.y;
    stu[c4 + 2][n] = v.z; stu[c4 + 3][n] = v.w;
  }
  for (int j = tid; j < NC; j += 128) {
#pragma unroll
    for (int n = 9; n < 16; ++n) stu[j][n] = 0.f;
  }

  // logits (same as K2a phase 1), written transposed into afft[n][l]
  v8f acc = {0.f, 0.f, 0.f, 0.f, 0.f, 0.f, 0.f, 0.f};
  int m0 = wave * 16;
  for (int ch = 0; ch < NCHUNK; ++ch) {
    __syncthreads();
    stage_pix_chunk(x, pixT, b, p, q, ch, tid);
    __syncthreads();
    for (int k0 = 0; k0 < CHUNK; k0 += 4) {
      int kk = k0 + 2 * half;
      v2f a;  a.x = pixT[kk][m0 + l15];      a.y = pixT[kk + 1][m0 + l15];
      int kg = ch * CHUNK + kk;
      v2f bb; bb.x = stu[kg][l15];           bb.y = stu[kg + 1][l15];
      acc = wmma4(a, bb, acc);
    }
  }
#pragma unroll
  for (int r = 0; r < 8; ++r) afft[l15][m0 + 8 * half + r] = acc[r];
  __syncthreads();

  if (tid < NL) {                        // softmax per pixel; zero pad rows 9..15
    float v[9]; float mx = -1e30f;
#pragma unroll
    for (int n = 0; n < 9; ++n) { v[n] = afft[n][tid]; mx = fmaxf(mx, v[n]); }
    float s = 0.f;
#pragma unroll
    for (int n = 0; n < 9; ++n) { v[n] = __expf((v[n] - mx) * SCALEF); s += v[n]; }
    float inv = 1.f / s;
#pragma unroll
    for (int n = 0; n < 9; ++n) afft[n][tid] = v[n] * inv;
#pragma unroll
    for (int n = 9; n < 16; ++n) afft[n][tid] = 0.f;
  }
  __syncthreads();

  // out tiles: 24 channel-tiles x 4 pixel-tiles, K = 16 (4 WMMA steps)
  for (int j = wave; j < 24 * 4; j += 4) {
    int ct = j >> 2, lt = j & 3;
    int c0 = ct * 16, l0 = lt * 16;
    v8f a3 = {0.f, 0.f, 0.f, 0.f, 0.f, 0.f, 0.f, 0.f};
#pragma unroll
    for (int k0 = 0; k0 < 16; k0 += 4) {
      int kk = k0 + 2 * half;
      v2f a;  a.x = stu[c0 + l15][kk];       a.y = stu[c0 + l15][kk + 1];
      v2f bb; bb.x = afft[kk][l0 + l15];     bb.y = afft[kk + 1][l0 + l15];
      a3 = wmma4(a, bb, a3);
    }
    int l = l0 + l15;
    int h = p * HP + (l >> 3);
    int w = q * WP + (l & 7);
    size_t base = ((size_t)(b * NC + c0 + 8 * half) * NH + h) * NW + w;
#pragma unroll
    for (int r = 0; r < 8; ++r) out[base + (size_t)r * NH * NW] = a3[r];
  }
}

// ---------------------------------------------------------------------------
extern "C" void kernel_launch(void* const* d_in, const int* in_sizes, int n_in,
                              void* d_out, int out_size, void* d_ws, size_t ws_size,
                              hipStream_t stream) {
  (void)in_sizes; (void)n_in; (void)out_size; (void)ws_size;
  const float* x  = (const float*)d_in[0];
  float* out      = (float*)d_out;
  float* ws       = (float*)d_ws;

  // workspace layout (floats): ~53 MB total
  size_t o = 0;
  float* st0    = ws + o; o += (size_t)NB * NM * NC;       // (B,28,28,C)
  float* st1    = ws + o; o += (size_t)NB * NM * NC;       // (B,28,28,C)
  float* st_ws  = ws + o; o += (size_t)NB * NM * 9 * NC;   // (B,M,9,C)
  float* affsum = ws + o; o += (size_t)NB * NM * 9;        // (B,M,9)
  float* normb  = ws + o;                                  // (B,M)

  k_avgpool<<<NB * NM, 128, 0, stream>>>(x, st0);
  k_iter0  <<<NB * NM, 128, 0, stream>>>(x, st0, st_ws, affsum);
  k_norm   <<<(NB * NM + 255) / 256, 256, 0, stream>>>(affsum, normb);
  k_fold   <<<(NB * NM * NC) / 256, 256, 0, stream>>>(st_ws, normb, st1);
  k_final  <<<NB * NM, 128, 0, stream>>>(x, st1, out);
}